// QHadamard_40243843564239
// MI455X (gfx1250) — compile-verified
//
#include <hip/hip_runtime.h>

typedef __attribute__((ext_vector_type(2))) float v2f;
typedef __attribute__((ext_vector_type(8))) float v8f;

#define FWHT_DIM 4096
#define WPB 2                 // waves per block (wave32); 2 * 16KB LDS = 32KB/block
#define TPB (WPB * 32)

#if __has_builtin(__builtin_amdgcn_tensor_load_to_lds) && \
    __has_builtin(__builtin_amdgcn_s_wait_tensorcnt)
#define USE_TDM 1
typedef unsigned int u32x4 __attribute__((ext_vector_type(4)));
typedef int          i32x4 __attribute__((ext_vector_type(4)));
typedef int          i32x8 __attribute__((ext_vector_type(8)));
#endif

// Scaled Sylvester H16 entry: (1/4)*(-1)^popcount(a&b). Three passes give the
// reference's HAD_SCALE = 1/64 exactly (powers of two: numerically exact fold).
__device__ __forceinline__ float hadq(int a, int b) {
    return (__popc(a & b) & 1) ? -0.25f : 0.25f;
}

__global__ __launch_bounds__(TPB) void fwht4096_wmma_kernel(
        const float* __restrict__ x, float* __restrict__ out, int nrows)
{
    __shared__ __align__(16) float lds[WPB * FWHT_DIM];  // single LDS object -> offset 0
    const int wave = threadIdx.x >> 5;
    const int lane = threadIdx.x & 31;
    const int row  = blockIdx.x * WPB + wave;
    if (row >= nrows) return;          // uniform per wave; EXEC stays all-1s

    float* L = lds + wave * FWHT_DIM;  // wave-private 16KB row buffer: no barriers
    const float* __restrict__ src = x   + (size_t)row * FWHT_DIM;
    float*       __restrict__ dst = out + (size_t)row * FWHT_DIM;

    // ---- Stage the row into LDS ----
#ifdef USE_TDM
    {
        // Tensor DMA descriptor (ISA ch.8): 1 row of 4096 f32, contiguous.
        const unsigned lds_byte = (unsigned)(wave * FWHT_DIM * 4);
        const unsigned long long ga = (unsigned long long)(__UINTPTR_TYPE__)src;
        u32x4 g0;
        g0[0] = 1u;                                   // count=1 (valid), user mode
        g0[1] = lds_byte;                             // lds_addr (bytes)
        g0[2] = (unsigned)(ga & 0xFFFFFFFFu);         // global_addr[31:0]
        g0[3] = (unsigned)((ga >> 32) & 0x01FFFFFFu)  // global_addr[56:32]
              | (2u << 30);                           // type=2 ("image")
        i32x8 g1;
        g1[0] = (2 << 16);                            // wg_mask=0, data_size=2 (4B)
        g1[1] = (int)((FWHT_DIM & 0xFFFFu) << 16);    // tensor_dim0[15:0] @ [63:48]
        g1[2] = (int)(1u << 16);                      // tensor_dim0[31:16]=0, tensor_dim1=1
        g1[3] = (int)((FWHT_DIM & 0xFFFFu) << 16);    // tile_dim0=4096 @ [127:112]
        g1[4] = 1;                                    // tile_dim1=1, tile_dim2=0
        g1[5] = FWHT_DIM;                             // tensor_dim0_stride[31:0]
        g1[6] = 0;                                    // stride[47:32]=0, dim1_stride lo=0
        g1[7] = 0;
        i32x4 g2 = {0, 0, 0, 0};                      // dims 2-4 unused
        i32x4 g3 = {0, 0, 0, 0};
        i32x8 g4 = {0, 0, 0, 0, 0, 0, 0, 0};          // 6-arg toolchain extra group: unused
        __builtin_amdgcn_tensor_load_to_lds(g0, g1, g2, g3, g4, 0);
        __builtin_amdgcn_s_wait_tensorcnt(0);         // 16KB now resident in LDS
        __asm__ volatile("" ::: "memory");            // keep LDS reads below the wait
    }
#else
#pragma unroll
    for (int i = 0; i < 32; ++i) {
        const int idx = (i * 32 + lane) * 4;
        *(float4*)(L + idx) = *(const float4*)(src + idx);
    }
#endif

    const int hi = lane >> 4;   // 0: lanes 0-15, 1: lanes 16-31
    const int lo = lane & 15;

    // Hoisted Hadamard fragments (H16 symmetric => identical as A- or B-operand):
    //   pass1 B frag (K=k0+r, N=lo)  ==  pass2/3 A frag (M=lo, K=k0+r)
    v2f h[4];
#pragma unroll
    for (int kk = 0; kk < 4; ++kk) {
        const int k0 = 4 * kk + 2 * hi;
        h[kk].x = hadq(k0 + 0, lo);
        h[kk].y = hadq(k0 + 1, lo);
    }

    // Fragment layouts (ISA 7.12.2, wave32, f32):
    //   A (16x4):  reg r holds (M=lo, K=4*kk + 2*hi + r)
    //   B (4x16):  reg r holds (K=4*kk + 2*hi + r, N=lo)
    //   C/D(16x16):reg r holds (M=r + 8*hi, N=lo)

    // ======== Pass 1: along k (fastest axis): D_t = X_t(16x16) * H, tile t=i, rows j ====
#pragma unroll
    for (int t = 0; t < 16; ++t) {
        v8f c = {};
#pragma unroll
        for (int kk = 0; kk < 4; ++kk) {
            const int k0 = 4 * kk + 2 * hi;
            v2f a;
            a.x = L[t * 256 + lo * 16 + (k0 + 0)];   // A[M=lo][K]
            a.y = L[t * 256 + lo * 16 + (k0 + 1)];
            c = __builtin_amdgcn_wmma_f32_16x16x4_f32(
                    false, a, false, h[kk], (short)0, c, false, false);
        }
#pragma unroll
        for (int r = 0; r < 8; ++r)                  // D[M=r+8hi][N=lo]
            L[t * 256 + (r + 8 * hi) * 16 + lo] = c[r];
    }

    // ======== Pass 2: along j: D_i = H * X_i(16x16), tile i ========
#pragma unroll
    for (int t = 0; t < 16; ++t) {
        v8f c = {};
#pragma unroll
        for (int kk = 0; kk < 4; ++kk) {
            const int k0 = 4 * kk + 2 * hi;
            v2f b;
            b.x = L[t * 256 + (k0 + 0) * 16 + lo];   // B[K=j][N=lo] = x[i=t][j][lo]
            b.y = L[t * 256 + (k0 + 1) * 16 + lo];
            c = __builtin_amdgcn_wmma_f32_16x16x4_f32(
                    false, h[kk], false, b, (short)0, c, false, false);
        }
#pragma unroll
        for (int r = 0; r < 8; ++r)
            L[t * 256 + (r + 8 * hi) * 16 + lo] = c[r];
    }

    // ======== Pass 3: along i: X is 16x256; tile = column block cb ========
#pragma unroll
    for (int cb = 0; cb < 16; ++cb) {
        v8f c = {};
#pragma unroll
        for (int kk = 0; kk < 4; ++kk) {
            const int k0 = 4 * kk + 2 * hi;
            v2f b;
            b.x = L[(k0 + 0) * 256 + cb * 16 + lo];  // B[K=i][N=lo] = x[i][16cb+lo]
            b.y = L[(k0 + 1) * 256 + cb * 16 + lo];
            c = __builtin_amdgcn_wmma_f32_16x16x4_f32(
                    false, h[kk], false, b, (short)0, c, false, false);
        }
#pragma unroll
        for (int r = 0; r < 8; ++r)
            L[(r + 8 * hi) * 256 + cb * 16 + lo] = c[r];
    }

    // ---- Store (scale already folded into H): coalesced b128 copies ----
#pragma unroll
    for (int i = 0; i < 32; ++i) {
        const int idx = (i * 32 + lane) * 4;
        *(float4*)(dst + idx) = *(const float4*)(L + idx);
    }
}

extern "C" void kernel_launch(void* const* d_in, const int* in_sizes, int n_in,
                              void* d_out, int out_size, void* d_ws, size_t ws_size,
                              hipStream_t stream) {
    (void)n_in; (void)d_ws; (void)ws_size; (void)out_size;
    const float* x = (const float*)d_in[0];
    float* out = (float*)d_out;
    const int nrows = in_sizes[0] / FWHT_DIM;          // 8192 for the reference shapes
    const int blocks = (nrows + WPB - 1) / WPB;
    fwht4096_wmma_kernel<<<blocks, TPB, 0, stream>>>(x, out, nrows);
}